// GATEncoder_36017595744480
// MI455X (gfx1250) — compile-verified
//
#include <hip/hip_runtime.h>
#include <math.h>

typedef __attribute__((ext_vector_type(2))) float v2f;
typedef __attribute__((ext_vector_type(8))) float v8f;

#define NEG_SLOPE 0.2f
#define BN_EPS 1e-5f

// ---------------------------------------------------------------------------
// float atomic max via signed/unsigned integer ordering trick
// (lowers to native GLOBAL_ATOMIC_MAX_I32 / GLOBAL_ATOMIC_MIN_U32 on gfx1250)
// ---------------------------------------------------------------------------
__device__ __forceinline__ void atomic_max_float(float* addr, float val) {
  if (val >= 0.0f) {
    atomicMax((int*)addr, __float_as_int(val));
  } else {
    atomicMin((unsigned int*)addr, __float_as_uint(val));
  }
}

// ---------------------------------------------------------------------------
// generic fill
// ---------------------------------------------------------------------------
__global__ void fill_kernel(float* __restrict__ p, float v, int n) {
  int i = blockIdx.x * blockDim.x + threadIdx.x;
  if (i < n) p[i] = v;
}

// ---------------------------------------------------------------------------
// C[nrows, ncols] = A[nrows, K] @ B[K, ncols], all f32, full precision.
// One wave per 16-row panel: A panel (16xK) is register-cached in WMMA
// A-operand layout (K/4 v2f's = K/2 VGPRs), then the wave sweeps all 16-col
// tiles with V_WMMA_F32_16X16X4_F32.  K is compile-time (128 here).
//
// A-operand layout (16x4 f32, 2 VGPRs per step):
//   lanes 0-15 : M=lane,    {K=k+0, K=k+1}
//   lanes 16-31: M=lane-16, {K=k+2, K=k+3}
// B-operand layout (4x16 f32): mirrored over N.
// C/D layout (8 VGPRs): vgpr v -> lanes 0-15: M=v, N=lane; lanes 16-31: M=v+8.
// ---------------------------------------------------------------------------
template <int K>
__global__ __launch_bounds__(32) void gemm_wmma_f32(
    const float* __restrict__ A, const float* __restrict__ B,
    float* __restrict__ C, int ncols) {
  const int lane = threadIdx.x;   // 0..31, EXEC all ones
  const int half = lane >> 4;     // 0 or 1
  const int l16  = lane & 15;
  const int row  = blockIdx.x * 16 + l16;
  const int koff = half * 2;

  // register-cache the A panel (aligned 8-byte loads)
  const float* __restrict__ Arow = A + (size_t)row * K;
  v2f a[K / 4];
#pragma unroll
  for (int ks = 0; ks < K / 4; ++ks) {
    a[ks] = *(const v2f*)(Arow + 4 * ks + koff);
  }

  const int m_base = blockIdx.x * 16 + half * 8;
  for (int nt = 0; nt < ncols; nt += 16) {
    const int col = nt + l16;
    v8f c = {};
#pragma unroll
    for (int ks = 0; ks < K / 4; ++ks) {
      v2f b;
      b.x = B[(size_t)(4 * ks + koff) * ncols + col];
      b.y = B[(size_t)(4 * ks + koff + 1) * ncols + col];
      c = __builtin_amdgcn_wmma_f32_16x16x4_f32(
          /*neg_a=*/false, a[ks], /*neg_b=*/false, b,
          /*c_mod=*/(short)0, c, /*reuse_a=*/false, /*reuse_b=*/false);
    }
#pragma unroll
    for (int v = 0; v < 8; ++v) {
      C[(size_t)(m_base + v) * ncols + nt + l16] = c[v];
    }
  }
}

// ---------------------------------------------------------------------------
// per-(node, head) attention coefficients: a_s = <h_row, att_src>, a_d likewise
// ---------------------------------------------------------------------------
__global__ void attn_kernel(const float* __restrict__ h,
                            const float* __restrict__ att_s,
                            const float* __restrict__ att_d,
                            float* __restrict__ a_s, float* __restrict__ a_d,
                            int n_nodes, int heads, int ch) {
  int i = blockIdx.x * blockDim.x + threadIdx.x;
  if (i >= n_nodes * heads) return;
  int hd = i % heads;
  const float* __restrict__ row = h + (size_t)(i / heads) * heads * ch + hd * ch;
  const float* __restrict__ as = att_s + hd * ch;
  const float* __restrict__ ad = att_d + hd * ch;
  float ss = 0.0f, sd = 0.0f;
  for (int c = 0; c < ch; ++c) {
    float v = row[c];
    ss += v * as[c];
    sd += v * ad[c];
  }
  a_s[i] = ss;
  a_d[i] = sd;
}

// ---------------------------------------------------------------------------
// pass 1 over edges: segment max of leaky_relu(a_src[s] + a_dst[d])
// edges e in [0,E) come from edge_index; e in [E, E+N) are self loops.
// ---------------------------------------------------------------------------
__global__ void edge_max_kernel(const long long* __restrict__ src_idx,
                                const long long* __restrict__ dst_idx,
                                int E, int N,
                                const float* __restrict__ a_s,
                                const float* __restrict__ a_d,
                                float* __restrict__ m, int heads) {
  int i = blockIdx.x * blockDim.x + threadIdx.x;
  int tot = (E + N) * heads;
  if (i >= tot) return;
  int e  = i / heads;
  int hd = i - e * heads;
  int s  = (e < E) ? (int)src_idx[e] : (e - E);
  int d  = (e < E) ? (int)dst_idx[e] : (e - E);
  float v = a_s[s * heads + hd] + a_d[d * heads + hd];
  v = (v > 0.0f) ? v : NEG_SLOPE * v;
  atomic_max_float(&m[d * heads + hd], v);
}

// ---------------------------------------------------------------------------
// pass 2: p = exp(e - m[dst]); segment-sum p; stash p for pass 3
// ---------------------------------------------------------------------------
__global__ void edge_expsum_kernel(const long long* __restrict__ src_idx,
                                   const long long* __restrict__ dst_idx,
                                   int E, int N,
                                   const float* __restrict__ a_s,
                                   const float* __restrict__ a_d,
                                   const float* __restrict__ m,
                                   float* __restrict__ p,
                                   float* __restrict__ ssum, int heads) {
  int i = blockIdx.x * blockDim.x + threadIdx.x;
  int tot = (E + N) * heads;
  if (i >= tot) return;
  int e  = i / heads;
  int hd = i - e * heads;
  int s  = (e < E) ? (int)src_idx[e] : (e - E);
  int d  = (e < E) ? (int)dst_idx[e] : (e - E);
  float v = a_s[s * heads + hd] + a_d[d * heads + hd];
  v = (v > 0.0f) ? v : NEG_SLOPE * v;
  float pe = __expf(v - m[d * heads + hd]);
  p[i] = pe;
  atomicAdd(&ssum[d * heads + hd], pe);
}

// ---------------------------------------------------------------------------
// pass 3: agg[dst] += alpha * h[src], vectorized: each thread handles 4
// contiguous channels (global_load_b128 gather + 4x f32 atomic adds).
// blockDim = 128; threads-per-edge = (heads*ch)/4 = 1 << tpe_shift;
// edges-per-block = 128 >> tpe_shift.
// ---------------------------------------------------------------------------
__global__ void edge_aggr_kernel(const long long* __restrict__ src_idx,
                                 const long long* __restrict__ dst_idx,
                                 int E, int N,
                                 const float* __restrict__ p,
                                 const float* __restrict__ ssum,
                                 const float* __restrict__ h,
                                 float* __restrict__ agg,
                                 int heads, int ch, int tpe_shift) {
  int e = blockIdx.x * (blockDim.x >> tpe_shift) + (threadIdx.x >> tpe_shift);
  if (e >= E + N) return;
  int t  = threadIdx.x & ((1 << tpe_shift) - 1);  // which float4 of the row
  int c0 = t * 4;
  int hd = c0 / ch;
  int s  = (e < E) ? (int)src_idx[e] : (e - E);
  int d  = (e < E) ? (int)dst_idx[e] : (e - E);
  float alpha = p[e * heads + hd] / (ssum[d * heads + hd] + 1e-16f);
  const float4 hv = *(const float4*)(h + (size_t)s * heads * ch + c0);
  float* ap = agg + (size_t)d * heads * ch + c0;
  atomicAdd(ap + 0, alpha * hv.x);
  atomicAdd(ap + 1, alpha * hv.y);
  atomicAdd(ap + 2, alpha * hv.z);
  atomicAdd(ap + 3, alpha * hv.w);
}

// ---------------------------------------------------------------------------
// bias + BatchNorm(eval) + ELU + residual, in place on agg1 (becomes layer-2
// input)
// ---------------------------------------------------------------------------
__global__ void bn_elu_res_kernel(float* __restrict__ a,
                                  const float* __restrict__ bias,
                                  const float* __restrict__ gamma,
                                  const float* __restrict__ beta,
                                  const float* __restrict__ mean,
                                  const float* __restrict__ var,
                                  const float* __restrict__ x,
                                  int C, int n) {
  int i = blockIdx.x * blockDim.x + threadIdx.x;
  if (i >= n) return;
  int c = i % C;
  float o = a[i] + bias[c];
  o = (o - mean[c]) * rsqrtf(var[c] + BN_EPS) * gamma[c] + beta[c];
  o = (o > 0.0f) ? o : (__expf(o) - 1.0f);   // ELU, alpha = 1
  a[i] = o + x[i];
}

// ---------------------------------------------------------------------------
// out += bias (final layer-2 bias; out was used as aggregation accumulator)
// ---------------------------------------------------------------------------
__global__ void add_bias_kernel(float* __restrict__ out,
                                const float* __restrict__ bias,
                                int C, int n) {
  int i = blockIdx.x * blockDim.x + threadIdx.x;
  if (i >= n) return;
  out[i] += bias[i % C];
}

// ---------------------------------------------------------------------------
// launch
// ---------------------------------------------------------------------------
extern "C" void kernel_launch(void* const* d_in, const int* in_sizes, int n_in,
                              void* d_out, int out_size, void* d_ws, size_t ws_size,
                              hipStream_t stream) {
  const int IN_C = 128, HEADS = 4, HID = 32, OUT_C = 64;

  const float*     x       = (const float*)d_in[0];
  const long long* eidx    = (const long long*)d_in[1];   // int64 [2, E]
  const float*     W1      = (const float*)d_in[2];       // [128, 128]
  const float*     attS1   = (const float*)d_in[3];       // [1,4,32]
  const float*     attD1   = (const float*)d_in[4];
  const float*     b1      = (const float*)d_in[5];
  const float*     gamma   = (const float*)d_in[6];
  const float*     beta    = (const float*)d_in[7];
  const float*     mean    = (const float*)d_in[8];
  const float*     var     = (const float*)d_in[9];
  const float*     W2      = (const float*)d_in[10];      // [128, 64]
  const float*     attS2   = (const float*)d_in[11];      // [1,1,64]
  const float*     attD2   = (const float*)d_in[12];
  const float*     b2      = (const float*)d_in[13];

  const int N    = in_sizes[0] / IN_C;     // 50000
  const int E    = in_sizes[1] / 2;        // 800000
  const int Etot = E + N;                  // self loops appended
  const long long* src_idx = eidx;
  const long long* dst_idx = eidx + E;

  float* out = (float*)d_out;              // [N, 64]

  // ---- workspace carve-up (floats) ----
  float* ws = (float*)d_ws;
  size_t o = 0;
  float* h1     = ws + o; o += (size_t)N * 128;   // x @ W1
  float* agg1   = ws + o; o += (size_t)N * 128;   // layer-1 aggregate -> layer-2 input
  float* h2     = ws + o; o += (size_t)N * 64;    // h @ W2
  float* a_src1 = ws + o; o += (size_t)N * 4;
  float* a_dst1 = ws + o; o += (size_t)N * 4;
  float* m1     = ws + o; o += (size_t)N * 4;
  float* s1     = ws + o; o += (size_t)N * 4;
  float* a_src2 = ws + o; o += (size_t)N;
  float* a_dst2 = ws + o; o += (size_t)N;
  float* m2     = ws + o; o += (size_t)N;
  float* s2     = ws + o; o += (size_t)N;
  float* p1     = ws + o; o += (size_t)Etot * 4;
  float* p2     = ws + o; o += (size_t)Etot;
  (void)ws_size; (void)n_in; (void)out_size;

  const float NEG_INF = -__builtin_huge_valf();
  auto blocks = [](long long n, int b) { return (int)((n + b - 1) / b); };

  // ---- init accumulators ----
  fill_kernel<<<blocks(N * 4, 256), 256, 0, stream>>>(m1, NEG_INF, N * 4);
  fill_kernel<<<blocks(N * 4, 256), 256, 0, stream>>>(s1, 0.0f, N * 4);
  fill_kernel<<<blocks((long long)N * 128, 256), 256, 0, stream>>>(agg1, 0.0f, N * 128);
  fill_kernel<<<blocks(N, 256), 256, 0, stream>>>(m2, NEG_INF, N);
  fill_kernel<<<blocks(N, 256), 256, 0, stream>>>(s2, 0.0f, N);
  fill_kernel<<<blocks((long long)N * 64, 256), 256, 0, stream>>>(out, 0.0f, N * 64);

  // ---- layer 1: GATConv(128 -> 4x32, concat) ----
  gemm_wmma_f32<128><<<N / 16, 32, 0, stream>>>(x, W1, h1, HEADS * HID);
  attn_kernel<<<blocks((long long)N * HEADS, 256), 256, 0, stream>>>(
      h1, attS1, attD1, a_src1, a_dst1, N, HEADS, HID);
  edge_max_kernel<<<blocks((long long)Etot * HEADS, 256), 256, 0, stream>>>(
      src_idx, dst_idx, E, N, a_src1, a_dst1, m1, HEADS);
  edge_expsum_kernel<<<blocks((long long)Etot * HEADS, 256), 256, 0, stream>>>(
      src_idx, dst_idx, E, N, a_src1, a_dst1, m1, p1, s1, HEADS);
  // 128 ch/row -> 32 threads/edge (shift 5), 4 edges per 128-thread block
  edge_aggr_kernel<<<blocks(Etot, 4), 128, 0, stream>>>(
      src_idx, dst_idx, E, N, p1, s1, h1, agg1, HEADS, HID, 5);

  // ---- BN(eval) + ELU + residual (in place, agg1 -> layer-2 input) ----
  bn_elu_res_kernel<<<blocks((long long)N * 128, 256), 256, 0, stream>>>(
      agg1, b1, gamma, beta, mean, var, x, 128, N * 128);

  // ---- layer 2: GATConv(128 -> 64, 1 head, concat=False == identity mean) ----
  gemm_wmma_f32<128><<<N / 16, 32, 0, stream>>>(agg1, W2, h2, OUT_C);
  attn_kernel<<<blocks(N, 256), 256, 0, stream>>>(
      h2, attS2, attD2, a_src2, a_dst2, N, 1, OUT_C);
  edge_max_kernel<<<blocks(Etot, 256), 256, 0, stream>>>(
      src_idx, dst_idx, E, N, a_src2, a_dst2, m2, 1);
  edge_expsum_kernel<<<blocks(Etot, 256), 256, 0, stream>>>(
      src_idx, dst_idx, E, N, a_src2, a_dst2, m2, p2, s2, 1);
  // 64 ch/row -> 16 threads/edge (shift 4), 8 edges per 128-thread block
  edge_aggr_kernel<<<blocks(Etot, 8), 128, 0, stream>>>(
      src_idx, dst_idx, E, N, p2, s2, h2, out, 1, OUT_C, 4);
  add_bias_kernel<<<blocks((long long)N * 64, 256), 256, 0, stream>>>(
      out, b2, OUT_C, N * 64);
}